// GATLayer_15333033247246
// MI455X (gfx1250) — compile-verified
//
#include <hip/hip_runtime.h>
#include <hip/hip_bf16.h>

typedef __attribute__((ext_vector_type(2))) float v2f;
typedef __attribute__((ext_vector_type(8))) float v8f;

#define HIDDEN 128

// --- Kernel 1: build padded A-matrix G (16x128: row0=u=W^T a1, row1=v=W^T a2,
//     rows 2..15 = 0) and c = (a1+a2)·b.  One 128-thread block. ---
__global__ void gat_prep_uv(const float* __restrict__ W, const float* __restrict__ b,
                            const float* __restrict__ att_w,
                            float* __restrict__ G, float* __restrict__ cs) {
  int j = threadIdx.x; // column 0..127
  float su = 0.f, sv = 0.f;
  for (int k = 0; k < HIDDEN; ++k) {
    float w = W[k * HIDDEN + j];
    su = fmaf(att_w[k], w, su);
    sv = fmaf(att_w[HIDDEN + k], w, sv);
  }
  G[j] = su;            // row 0: u
  G[HIDDEN + j] = sv;   // row 1: v
#pragma unroll
  for (int r = 2; r < 16; ++r) G[r * HIDDEN + j] = 0.f;  // zero padding rows
  __shared__ float tb[HIDDEN];
  tb[j] = (att_w[j] + att_w[HIDDEN + j]) * b[j];
  __syncthreads();
  if (j == 0) {
    float s = 0.f;
    for (int k = 0; k < HIDDEN; ++k) s += tb[k];
    cs[0] = s;
  }
}

// --- Kernel 2: zero the edge-exp accumulator (must happen every call) ---
__global__ void gat_zero(float* __restrict__ nacc, int n) {
  int i = blockIdx.x * blockDim.x + threadIdx.x;
  if (i < n) nacc[i] = 0.f;
}

// --- Kernel 3: p[i] = u·x[i], q[i] = v·x[i] via V_WMMA_F32_16X16X4_F32 ---
// One wave32 per TWO 16-node tiles (shared A operand, 2 accumulator chains).
// A layout: lane%16 = M, VGPR j holds K = (lane>=16?2:0)+j.
// B layout: lane%16 = N, VGPR j holds K = (lane>=16?2:0)+j.
// C layout: VGPR r = row M=r (lanes 0-15), so acc[0]=p, acc[1]=q on lanes 0-15.
__global__ void gat_pq_wmma(const float* __restrict__ x, const float* __restrict__ G,
                            float* __restrict__ p, float* __restrict__ q,
                            int n_tiles, int n_nodes) {
  int lane = threadIdx.x & 31;
  int wv   = (blockIdx.x * blockDim.x + threadIdx.x) >> 5; // global wave id
  int t0 = wv * 2;
  int t1 = t0 + 1;
  if (t0 >= n_tiles) return;                               // wave-uniform exit
  bool has1 = (t1 < n_tiles);

  int m    = lane & 15;
  int koff = (lane >> 4) << 1;                             // 0 or 2
  int n0 = t0 * 16 + m; if (n0 >= n_nodes) n0 = n_nodes - 1;
  int n1 = has1 ? (t1 * 16 + m) : n0; if (n1 >= n_nodes) n1 = n_nodes - 1;

  const float* a_ptr = G + (size_t)m  * HIDDEN + koff;     // padded: no branch
  const float* b0ptr = x + (size_t)n0 * HIDDEN + koff;
  const float* b1ptr = x + (size_t)n1 * HIDDEN + koff;

  v8f acc0 = {0.f, 0.f, 0.f, 0.f, 0.f, 0.f, 0.f, 0.f};
  v8f acc1 = {0.f, 0.f, 0.f, 0.f, 0.f, 0.f, 0.f, 0.f};
#pragma unroll
  for (int k0 = 0; k0 < HIDDEN; k0 += 4) {
    v2f aa  = *(const v2f*)(a_ptr + k0);
    v2f bb0 = *(const v2f*)(b0ptr + k0);
    v2f bb1 = *(const v2f*)(b1ptr + k0);
    // (neg_a, A, neg_b, B, c_mod, C, reuse_a, reuse_b)
    acc0 = __builtin_amdgcn_wmma_f32_16x16x4_f32(false, aa, false, bb0,
                                                 (short)0, acc0, false, false);
    acc1 = __builtin_amdgcn_wmma_f32_16x16x4_f32(false, aa, false, bb1,
                                                 (short)0, acc1, false, false);
  }
  if (lane < 16) {
    int o0 = t0 * 16 + lane;
    if (o0 < n_nodes) { p[o0] = acc0[0]; q[o0] = acc0[1]; }
    int o1 = t1 * 16 + lane;
    if (has1 && o1 < n_nodes) { p[o1] = acc1[0]; q[o1] = acc1[1]; }
  }
}

// --- Kernel 4: per-edge attention scalar + segment-sum via f32 global atomics ---
__global__ void gat_edge(const int* __restrict__ e, const float* __restrict__ p,
                         const float* __restrict__ q, const float* __restrict__ cs,
                         float* __restrict__ nacc, int n_edges) {
  int i = blockIdx.x * blockDim.x + threadIdx.x;
  if (i >= n_edges) return;
  int srcn = e[2 * i];
  int dstn = e[2 * i + 1];
  float z = p[srcn] + q[dstn] + cs[0];
  float lz = z > 0.f ? z : 0.2f * z;       // leaky_relu, slope 0.2
  atomicAdd(&nacc[dstn], expf(lz));        // global_atomic_add_f32
}

// --- Kernel 5: h = relu(x/n'), y = h + x, RMS-norm. One wave32 per node. ---
__global__ void gat_out(const float* __restrict__ x, const float* __restrict__ nacc,
                        const float* __restrict__ nw, const float* __restrict__ nb,
                        float* __restrict__ out, int n_nodes) {
  int lane = threadIdx.x & 31;
  int node = (blockIdx.x * blockDim.x + threadIdx.x) >> 5;
  if (node >= n_nodes) return;
  float nv = nacc[node];
  float denom = (nv == 0.f) ? 1.f : nv;
  float4 xv = ((const float4*)(x + (size_t)node * HIDDEN))[lane];
  float4 y;
  y.x = fmaxf(xv.x / denom, 0.f) + xv.x;
  y.y = fmaxf(xv.y / denom, 0.f) + xv.y;
  y.z = fmaxf(xv.z / denom, 0.f) + xv.z;
  y.w = fmaxf(xv.w / denom, 0.f) + xv.w;
  float ss = y.x * y.x + y.y * y.y + y.z * y.z + y.w * y.w;
  ss += __shfl_xor(ss, 16, 32);
  ss += __shfl_xor(ss,  8, 32);
  ss += __shfl_xor(ss,  4, 32);
  ss += __shfl_xor(ss,  2, 32);
  ss += __shfl_xor(ss,  1, 32);
  float inv_rms = rsqrtf(ss * (1.0f / HIDDEN) + 1e-6f);
  int c = lane * 4;
  float4 o;
  o.x = nw[c + 0] * (y.x * inv_rms) + nb[c + 0];
  o.y = nw[c + 1] * (y.y * inv_rms) + nb[c + 1];
  o.z = nw[c + 2] * (y.z * inv_rms) + nb[c + 2];
  o.w = nw[c + 3] * (y.w * inv_rms) + nb[c + 3];
  ((float4*)(out + (size_t)node * HIDDEN))[lane] = o;
}

extern "C" void kernel_launch(void* const* d_in, const int* in_sizes, int n_in,
                              void* d_out, int out_size, void* d_ws, size_t ws_size,
                              hipStream_t stream) {
  const float* x     = (const float*)d_in[0];
  const float* W     = (const float*)d_in[1];
  const float* b     = (const float*)d_in[2];
  const float* att_w = (const float*)d_in[3];
  const float* nw    = (const float*)d_in[4];
  const float* nb    = (const float*)d_in[5];
  const int*   e     = (const int*)d_in[6];
  float* out = (float*)d_out;

  const int n_nodes = in_sizes[0] / HIDDEN;  // 100000
  const int n_edges = in_sizes[6] / 2;       // 640000

  // workspace layout (floats): [G: 16*128][cs: 16 pad][p: N][q: N][nacc: N]
  float* G    = (float*)d_ws;
  float* cs   = G + 16 * HIDDEN;
  float* p    = cs + 16;
  float* q    = p + n_nodes;
  float* nacc = q + n_nodes;

  gat_prep_uv<<<1, 128, 0, stream>>>(W, b, att_w, G, cs);
  gat_zero<<<(n_nodes + 255) / 256, 256, 0, stream>>>(nacc, n_nodes);

  int n_tiles = (n_nodes + 15) / 16;              // 16-node tiles
  int n_waves = (n_tiles + 1) / 2;                // one wave32 per 2 tiles
  int pq_blocks = (n_waves * 32 + 127) / 128;     // 4 waves per block
  gat_pq_wmma<<<pq_blocks, 128, 0, stream>>>(x, G, p, q, n_tiles, n_nodes);

  gat_edge<<<(n_edges + 255) / 256, 256, 0, stream>>>(e, p, q, cs, nacc, n_edges);

  gat_out<<<(n_nodes * 32 + 255) / 256, 256, 0, stream>>>(x, nacc, nw, nb, out, n_nodes);
}